// ParamModel_3204045603551
// MI455X (gfx1250) — compile-verified
//
#include <hip/hip_runtime.h>
#include <hip/hip_bf16.h>

typedef float v2f __attribute__((ext_vector_type(2)));
typedef float v8f __attribute__((ext_vector_type(8)));

constexpr int   Hdim   = 64;
constexpr int   Vdim   = 64;
constexpr int   Bsz    = 256;
constexpr int   Lseq   = 2048;
constexpr int   Tsteps = (Lseq - 2) / 2;   // 1023
constexpr float LRc    = 0.01f;

// ---------------------------------------------------------------------------
// Kernel 1: hs_table[v][i] = LayerNorm(e + FFN(e)) for each of the 64 tokens.
// One block of 64 threads; thread i owns feature dim i, loop over tokens.
// ---------------------------------------------------------------------------
__global__ __launch_bounds__(64) void k_build_table(
    const float* __restrict__ embed, const float* __restrict__ ff_w1,
    const float* __restrict__ ff_b1, const float* __restrict__ ff_w2,
    const float* __restrict__ ff_b2, const float* __restrict__ ln_g,
    const float* __restrict__ ln_b, float* __restrict__ tbl) {
  __shared__ float e[64];
  __shared__ float z[128];
  __shared__ float hbuf[64];
  const int i = threadIdx.x;  // 0..63
  for (int v = 0; v < Vdim; ++v) {
    e[i] = embed[v * 64 + i];
    __syncthreads();
    for (int q = 0; q < 2; ++q) {
      int j = i + 64 * q;
      float acc = ff_b1[j];
      for (int h = 0; h < 64; ++h) acc += e[h] * ff_w1[h * 128 + j];
      z[j] = fmaxf(acc, 0.0f);
    }
    __syncthreads();
    float acc = ff_b2[i];
    for (int j = 0; j < 128; ++j) acc += z[j] * ff_w2[j * 64 + i];
    hbuf[i] = e[i] + acc;
    __syncthreads();
    float mu = 0.0f;
    for (int h = 0; h < 64; ++h) mu += hbuf[h];
    mu *= (1.0f / 64.0f);
    float var = 0.0f;
    for (int h = 0; h < 64; ++h) { float d = hbuf[h] - mu; var += d * d; }
    var *= (1.0f / 64.0f);
    tbl[v * 64 + i] = (hbuf[i] - mu) / sqrtf(var + 1e-5f) * ln_g[i] + ln_b[i];
    __syncthreads();
  }
}

// ---------------------------------------------------------------------------
// Kernel 2: TTT inner scan. One wave32 per batch chain; W1/W2/b1/b2 + hs table
// resident in LDS. Rank-1 parameter updates done with V_WMMA_F32_16X16X4_F32
// (K=0 column/row populated only -> exact f32 outer-product accumulate).
// ---------------------------------------------------------------------------
__global__ __launch_bounds__(32) void k_ttt(
    const int* __restrict__ seq, const float* __restrict__ w1_in,
    const float* __restrict__ b1_in, const float* __restrict__ w2_in,
    const float* __restrict__ b2_in, const float* __restrict__ tbl_g,
    float* __restrict__ ctx_out) {
  __shared__ float tbl[64 * 64];   // hs table
  __shared__ float W1[64 * 16];    // row-major [h][j]
  __shared__ float W2[16 * 64];    // row-major [j][i]
  __shared__ float b1[16];
  __shared__ float b2[64];
  __shared__ float zpre[16];
  __shared__ float act[16];
  __shared__ float dout[64];
  __shared__ float da[16];
  __shared__ float red[32];

  const int t     = threadIdx.x;
  const int b     = blockIdx.x;
  const int jlane = t & 15;   // column within tile / inner index
  const int plane = t >> 4;   // half-wave id (also C-layout row-half select)

  for (int idx = t; idx < 64 * 64; idx += 32) tbl[idx] = tbl_g[idx];
  for (int idx = t; idx < 64 * 16; idx += 32) W1[idx] = w1_in[idx];
  for (int idx = t; idx < 16 * 64; idx += 32) W2[idx] = w2_in[idx];
  if (t < 16) b1[t] = b1_in[t];
  b2[t]      = b2_in[t];
  b2[t + 32] = b2_in[t + 32];
  __syncthreads();

  const int* sb = seq + b * Lseq;

  for (int s = 0; s < Tsteps; ++s) {
    const float* kv = &tbl[sb[2 * s] * 64];
    const float* vv = &tbl[sb[2 * s + 1] * 64];

    // z1[j] = k @ W1[:,j] + b1[j]  (half-wave partials + fold)
    float acc = 0.0f;
    for (int h = 32 * plane; h < 32 * plane + 32; ++h)
      acc += kv[h] * W1[h * 16 + jlane];
    red[t] = acc;
    __syncthreads();
    if (t < 16) {
      float z = red[t] + red[t + 16] + b1[t];
      zpre[t] = z;
      act[t]  = fmaxf(z, 0.0f);
    }
    __syncthreads();

    // out = a @ W2 + b2 ; dout = 2*(out - v)/H ; b2 -= LR*dout
    for (int q = 0; q < 2; ++q) {
      int i = t + 32 * q;
      float o = b2[i];
      for (int j = 0; j < 16; ++j) o += act[j] * W2[j * 64 + i];
      float d = (o - vv[i]) * (2.0f / 64.0f);
      dout[i] = d;
      b2[i]  -= LRc * d;
    }
    __syncthreads();

    // da[j] = W2[j,:] . dout   (must use pre-update W2)
    acc = 0.0f;
    for (int i = 32 * plane; i < 32 * plane + 32; ++i)
      acc += W2[jlane * 64 + i] * dout[i];
    red[t] = acc;
    __syncthreads();
    if (t < 16) {
      float dz = (zpre[t] > 0.0f) ? (red[t] + red[t + 16]) : 0.0f;
      da[t] = dz;                 // = dz1
      b1[t] -= LRc * dz;
    }
    __syncthreads();

    // W2 += (-LR * a) (x) dout : four 16x16 tiles, rank-1 WMMA each.
    {
      v2f av; av.x = (t < 16) ? (-LRc) * act[jlane] : 0.0f; av.y = 0.0f;
      for (int c = 0; c < 4; ++c) {
        v2f bv; bv.x = (t < 16) ? dout[16 * c + jlane] : 0.0f; bv.y = 0.0f;
        v8f cv;
        for (int g = 0; g < 8; ++g)
          cv[g] = W2[(g + 8 * plane) * 64 + 16 * c + jlane];
        cv = __builtin_amdgcn_wmma_f32_16x16x4_f32(false, av, false, bv,
                                                   (short)0, cv, false, false);
        for (int g = 0; g < 8; ++g)
          W2[(g + 8 * plane) * 64 + 16 * c + jlane] = cv[g];
      }
    }
    // W1 += (-LR * k) (x) dz1 : four 16x16 tiles, rank-1 WMMA each.
    {
      v2f bv; bv.x = (t < 16) ? da[jlane] : 0.0f; bv.y = 0.0f;
      for (int r = 0; r < 4; ++r) {
        v2f av; av.x = (t < 16) ? (-LRc) * kv[16 * r + jlane] : 0.0f; av.y = 0.0f;
        v8f cv;
        for (int g = 0; g < 8; ++g)
          cv[g] = W1[(16 * r + g + 8 * plane) * 16 + jlane];
        cv = __builtin_amdgcn_wmma_f32_16x16x4_f32(false, av, false, bv,
                                                   (short)0, cv, false, false);
        for (int g = 0; g < 8; ++g)
          W1[(16 * r + g + 8 * plane) * 16 + jlane] = cv[g];
      }
    }
    __syncthreads();
  }

  // ctx = mlp(p_final, hs[:, L-1])
  const float* qv = &tbl[sb[Lseq - 1] * 64];
  float acc = 0.0f;
  for (int h = 32 * plane; h < 32 * plane + 32; ++h)
    acc += qv[h] * W1[h * 16 + jlane];
  red[t] = acc;
  __syncthreads();
  if (t < 16) act[t] = fmaxf(red[t] + red[t + 16] + b1[t], 0.0f);
  __syncthreads();
  for (int q = 0; q < 2; ++q) {
    int i = t + 32 * q;
    float o = b2[i];
    for (int j = 0; j < 16; ++j) o += act[j] * W2[j * 64 + i];
    ctx_out[b * 64 + i] = o;
  }
}

// ---------------------------------------------------------------------------
// Kernel 3: out = ctx(256x64) @ out_w(64x64) + out_b, tiled 16x16 per wave,
// K=4 f32 WMMA accumulation chain (16 steps over K=64).
// A layout (ISA 7.12.2): lanes 0-15 V0/V1 = K0/K1, lanes 16-31 V0/V1 = K2/K3.
// B layout mirrored:      lanes 0-15 V0/V1 = K0/K1, lanes 16-31 V0/V1 = K2/K3.
// ---------------------------------------------------------------------------
__global__ __launch_bounds__(32) void k_outproj(
    const float* __restrict__ ctx, const float* __restrict__ out_w,
    const float* __restrict__ out_b, float* __restrict__ out) {
  const int t    = threadIdx.x;
  const int row0 = blockIdx.x * 16;   // 16 row tiles (256 rows)
  const int col0 = blockIdx.y * 16;   // 4 col tiles (64 cols)
  const int m    = t & 15;
  const int hh   = t >> 4;
  v8f acc = {0.f, 0.f, 0.f, 0.f, 0.f, 0.f, 0.f, 0.f};
  for (int kt = 0; kt < 16; ++kt) {
    int kb = 4 * kt + 2 * hh;
    v2f A;  A.x  = ctx[(row0 + m) * 64 + kb];
            A.y  = ctx[(row0 + m) * 64 + kb + 1];
    v2f Bv; Bv.x = out_w[kb * 64 + col0 + m];
            Bv.y = out_w[(kb + 1) * 64 + col0 + m];
    acc = __builtin_amdgcn_wmma_f32_16x16x4_f32(false, A, false, Bv,
                                                (short)0, acc, false, false);
  }
  float bias = out_b[col0 + m];
  for (int g = 0; g < 8; ++g) {
    int row = row0 + g + 8 * hh;
    out[row * 64 + col0 + m] = acc[g] + bias;
  }
}

// ---------------------------------------------------------------------------
extern "C" void kernel_launch(void* const* d_in, const int* in_sizes, int n_in,
                              void* d_out, int out_size, void* d_ws, size_t ws_size,
                              hipStream_t stream) {
  const int*   seq    = (const int*)  d_in[0];
  const float* embed  = (const float*)d_in[1];
  const float* ff_w1  = (const float*)d_in[2];
  const float* ff_b1  = (const float*)d_in[3];
  const float* ff_w2  = (const float*)d_in[4];
  const float* ff_b2  = (const float*)d_in[5];
  const float* ln_g   = (const float*)d_in[6];
  const float* ln_b   = (const float*)d_in[7];
  const float* w1     = (const float*)d_in[8];
  const float* b1     = (const float*)d_in[9];
  const float* w2     = (const float*)d_in[10];
  const float* b2     = (const float*)d_in[11];
  const float* out_w  = (const float*)d_in[12];
  const float* out_b  = (const float*)d_in[13];
  float* outp = (float*)d_out;

  float* tbl = (float*)d_ws;            // 64*64 floats
  float* ctx = tbl + 64 * 64;           // 256*64 floats

  k_build_table<<<1, 64, 0, stream>>>(embed, ff_w1, ff_b1, ff_w2, ff_b2,
                                      ln_g, ln_b, tbl);
  k_ttt<<<Bsz, 32, 0, stream>>>(seq, w1, b1, w2, b2, tbl, ctx);
  dim3 g3(16, 4);
  k_outproj<<<g3, 32, 0, stream>>>(ctx, out_w, out_b, outp);
}